// Decoder_12489764897128
// MI455X (gfx1250) — compile-verified
//
#include <hip/hip_runtime.h>

// ---------------------------------------------------------------------------
// 2-layer GCN for MI455X (gfx1250, wave32).
//  - Dense transforms use V_WMMA_F32_16X16X4_F32 (full f32 precision; GEMMs
//    are ~5 GFLOP total and far from the bottleneck, so no low-precision path).
//  - Edge aggregation is the memory-bound part; h (51.2 MB) and the
//    accumulator fit in the 192 MB L2, so gather + global_atomic_add_f32 run
//    at L2 bandwidth. One wave per edge, float4/float2 per lane (coalesced).
// ---------------------------------------------------------------------------

typedef __attribute__((ext_vector_type(2))) float v2f;
typedef __attribute__((ext_vector_type(8))) float v8f;

#define K_DIM 128  // input channels of both layers (IN_CH == HID == 128)

// ---------------------------------------------------------------- utilities
__global__ void fill_f32(float* __restrict__ p, float v, long n) {
    long i = (long)blockIdx.x * blockDim.x + threadIdx.x;
    if (i < n) p[i] = v;
}

__global__ void degree_kernel(const int* __restrict__ dst,
                              float* __restrict__ deg, int n_edges) {
    int e = blockIdx.x * blockDim.x + threadIdx.x;
    if (e < n_edges) atomicAdd(&deg[dst[e]], 1.0f);
}

__global__ void rsqrt_inplace(float* __restrict__ deg, int n) {
    int i = blockIdx.x * blockDim.x + threadIdx.x;
    if (i < n) deg[i] = rsqrtf(deg[i]);  // deg >= 1 (self-loop), no zero case
}

// ------------------------------------------------------------- WMMA GEMM
// C[nrows, NCOLS] = A[nrows, 128] @ W[128, NCOLS]
// One wave32 computes one 16x16 output tile; K-loop = 32 x v_wmma_f32_16x16x4_f32.
//
// f32 16x16x4 operand layout (ISA 7.12.2): for lane L,
//   A: M = L%16, VGPR v holds K = 2*(L/16) + v
//   B: N = L%16, VGPR v holds K = 2*(L/16) + v
//   D: VGPR r   holds row r + 8*(L/16), col L%16
template <int NCOLS>
__global__ void gemm_wmma_f32(const float* __restrict__ A,
                              const float* __restrict__ W,
                              float* __restrict__ C, int nrows) {
    constexpr int TC = NCOLS / 16;               // tiles along columns
    const int lane  = threadIdx.x & 31;
    const int wave  = threadIdx.x >> 5;
    const int tile  = blockIdx.x * (blockDim.x >> 5) + wave;
    const int trow  = tile / TC;
    const int tcol  = tile % TC;
    const int rbase = trow * 16;
    if (rbase >= nrows) return;

    const int m     = lane & 15;   // row-in-tile (A) / col-in-tile (B)
    const int khalf = lane >> 4;   // 0 or 1 -> K offset 0 or 2

    const int arow_idx = (rbase + m < nrows) ? (rbase + m) : (nrows - 1);
    const float* arow = A + (long)arow_idx * K_DIM;
    const float* wcol = W + tcol * 16 + m;

    v8f acc = {};
#pragma unroll 8
    for (int k = 0; k < K_DIM; k += 4) {
        const int kk = k + 2 * khalf;
        v2f a, b;
        a.x = arow[kk];
        a.y = arow[kk + 1];
        b.x = wcol[(long)kk * NCOLS];
        b.y = wcol[(long)(kk + 1) * NCOLS];
        // (neg_a, A, neg_b, B, c_mod, C, reuse_a, reuse_b)
        acc = __builtin_amdgcn_wmma_f32_16x16x4_f32(
            false, a, false, b, (short)0, acc, false, false);
    }

    // D store: fast path for full tiles (the only case when nrows % 16 == 0);
    // single whole-tile test instead of 8 per-row exec-mask branches.
    if (rbase + 16 <= nrows) {
        float* cbase = C + (long)(rbase + 8 * khalf) * NCOLS + tcol * 16 + m;
#pragma unroll
        for (int r = 0; r < 8; ++r) cbase[(long)r * NCOLS] = acc[r];
    } else {
#pragma unroll
        for (int r = 0; r < 8; ++r) {
            const int row = rbase + r + 8 * khalf;
            if (row < nrows) C[(long)row * NCOLS + tcol * 16 + m] = acc[r];
        }
    }
}

// ---------------------------------------------------- edge scatter-gather
// out[dst[e]] += h[src[e]] * dinv[src[e]] * dinv[dst[e]]
// One wave per edge; CH/32 contiguous floats per lane (coalesced row I/O).
template <int CH>
__global__ void aggregate_edges(const float* __restrict__ h,
                                const int* __restrict__ src,
                                const int* __restrict__ dst,
                                const float* __restrict__ dinv,
                                float* __restrict__ out, int n_edges) {
    const int lane = threadIdx.x & 31;
    const int e    = blockIdx.x * (blockDim.x >> 5) + (threadIdx.x >> 5);
    if (e >= n_edges) return;

    const int   s   = src[e];
    const int   d   = dst[e];
    const float nrm = dinv[s] * dinv[d];

    constexpr int V = CH / 32;  // 4 (CH=128) or 2 (CH=64) floats per lane
    const float* hp = h   + (long)s * CH + lane * V;
    float*       op = out + (long)d * CH + lane * V;

    float vals[V];
#pragma unroll
    for (int i = 0; i < V; ++i) vals[i] = hp[i];   // vectorizes to b128/b64 load
#pragma unroll
    for (int i = 0; i < V; ++i) atomicAdd(&op[i], vals[i] * nrm);
}

// ------------------------------------------------ epilogue: self-loop+bias+ReLU
// agg[i][c] = relu(agg[i][c] + h[i][c]*dinv[i]^2 + bias[c])
template <int CH>
__global__ void bias_relu_self(const float* __restrict__ h,
                               const float* __restrict__ dinv,
                               const float* __restrict__ bias,
                               float* __restrict__ agg, int n_nodes) {
    long idx = (long)blockIdx.x * blockDim.x + threadIdx.x;
    if (idx >= (long)n_nodes * CH) return;
    const int node = (int)(idx / CH);
    const int c    = (int)(idx % CH);
    const float di = dinv[node];
    const float v  = agg[idx] + h[idx] * di * di + bias[c];
    agg[idx] = v > 0.0f ? v : 0.0f;
}

// ---------------------------------------------------------------------------
extern "C" void kernel_launch(void* const* d_in, const int* in_sizes, int n_in,
                              void* d_out, int out_size, void* d_ws, size_t ws_size,
                              hipStream_t stream) {
    const float* x  = (const float*)d_in[0];
    const int*   ei = (const int*)d_in[1];   // [2, E] int32 (JAX default x64-off)
    const float* W1 = (const float*)d_in[2];
    const float* b1 = (const float*)d_in[3];
    const float* W2 = (const float*)d_in[4];
    const float* b2 = (const float*)d_in[5];

    const int N   = in_sizes[0] / K_DIM;   // 100000
    const int E   = in_sizes[1] / 2;       // 1600000
    (void)n_in; (void)ws_size; (void)out_size;

    const int* src = ei;
    const int* dst = ei + E;
    float* out = (float*)d_out;

    // workspace layout (256B aligned): deg/dinv | bufA (N*128) | bufB (N*128)
    char* ws = (char*)d_ws;
    const size_t degBytes = (((size_t)N * 4) + 255) & ~(size_t)255;
    float* deg  = (float*)ws;
    float* bufA = (float*)(ws + degBytes);
    float* bufB = (float*)(ws + degBytes + (size_t)N * 128 * 4);

    const int TB = 256;

    // ---- symmetric normalization: deg=1 (self-loop) + scatter, then rsqrt
    fill_f32<<<(N + TB - 1) / TB, TB, 0, stream>>>(deg, 1.0f, N);
    degree_kernel<<<(E + TB - 1) / TB, TB, 0, stream>>>(dst, deg, E);
    rsqrt_inplace<<<(N + TB - 1) / TB, TB, 0, stream>>>(deg, N);  // deg -> dinv

    // ---- layer 1: h1 = x @ W1 ; a1 = relu(scatter(h1*norm) + self + b1)
    {
        const int tiles = ((N + 15) / 16) * (128 / 16);
        gemm_wmma_f32<128><<<(tiles + 7) / 8, 256, 0, stream>>>(x, W1, bufA, N);
        const long n1 = (long)N * 128;
        fill_f32<<<(unsigned)((n1 + TB - 1) / TB), TB, 0, stream>>>(bufB, 0.0f, n1);
        aggregate_edges<128><<<(E + 7) / 8, 256, 0, stream>>>(bufA, src, dst, deg, bufB, E);
        bias_relu_self<128><<<(unsigned)((n1 + TB - 1) / TB), TB, 0, stream>>>(bufA, deg, b1, bufB, N);
    }

    // ---- layer 2: h2 = a1 @ W2 ; out = relu(scatter(h2*norm) + self + b2)
    {
        const int tiles = ((N + 15) / 16) * (64 / 16);
        gemm_wmma_f32<64><<<(tiles + 7) / 8, 256, 0, stream>>>(bufB, W2, bufA, N);
        const long n2 = (long)N * 64;
        fill_f32<<<(unsigned)((n2 + TB - 1) / TB), TB, 0, stream>>>(out, 0.0f, n2);
        aggregate_edges<64><<<(E + 7) / 8, 256, 0, stream>>>(bufA, src, dst, deg, out, E);
        bias_relu_self<64><<<(unsigned)((n2 + TB - 1) / TB), TB, 0, stream>>>(bufA, deg, b2, out, N);
    }
}